// QuantumRNNCell_20512763805780
// MI455X (gfx1250) — compile-verified
//
#include <hip/hip_runtime.h>
#include <hip/hip_bf16.h>

typedef float v2f __attribute__((ext_vector_type(2)));
typedef float v8f __attribute__((ext_vector_type(8)));

// ---------------------------------------------------------------------------
// Kernel 1: 6-qubit state-vector simulation, one batch element per thread.
// State = 64 complex amplitudes kept entirely in VGPRs via full unrolling.
// ---------------------------------------------------------------------------

__device__ __forceinline__ void rx_gate(float* sr, float* si, int q, float c, float s) {
    const int tb = 1 << (5 - q);
#pragma unroll
    for (int i0 = 0; i0 < 64; ++i0) {
        if (i0 & tb) continue;
        const int i1 = i0 | tb;
        float ar = sr[i0], ai = si[i0], br = sr[i1], bi = si[i1];
        // a' = c*a - i*s*b ; b' = -i*s*a + c*b
        sr[i0] = c * ar + s * bi;
        si[i0] = c * ai - s * br;
        sr[i1] = s * ai + c * br;
        si[i1] = -s * ar + c * bi;
    }
}

__device__ __forceinline__ void ry_gate(float* sr, float* si, int q, float c, float s) {
    const int tb = 1 << (5 - q);
#pragma unroll
    for (int i0 = 0; i0 < 64; ++i0) {
        if (i0 & tb) continue;
        const int i1 = i0 | tb;
        float ar = sr[i0], ai = si[i0], br = sr[i1], bi = si[i1];
        // a' = c*a - s*b ; b' = s*a + c*b   (real matrix)
        sr[i0] = c * ar - s * br;
        si[i0] = c * ai - s * bi;
        sr[i1] = s * ar + c * br;
        si[i1] = s * ai + c * bi;
    }
}

__device__ __forceinline__ void rz_gate(float* sr, float* si, int q, float c, float s) {
    const int tb = 1 << (5 - q);
#pragma unroll
    for (int i0 = 0; i0 < 64; ++i0) {
        if (i0 & tb) continue;
        const int i1 = i0 | tb;
        float ar = sr[i0], ai = si[i0], br = sr[i1], bi = si[i1];
        // a' = (c - i s)*a ; b' = (c + i s)*b
        sr[i0] = c * ar + s * ai;
        si[i0] = c * ai - s * ar;
        sr[i1] = c * br - s * bi;
        si[i1] = c * bi + s * br;
    }
}

__device__ __forceinline__ void cnot_gate(float* sr, float* si, int cq, int tq) {
    const int cb = 1 << (5 - cq);
    const int tb = 1 << (5 - tq);
#pragma unroll
    for (int i = 0; i < 64; ++i) {
        if ((i & cb) && !(i & tb)) {
            const int j = i | tb;
            float tr = sr[i], ti = si[i];
            sr[i] = sr[j]; si[i] = si[j];
            sr[j] = tr;    si[j] = ti;
        }
    }
}

__global__ __launch_bounds__(128, 1) void quantum_sim_kernel(
    const float* __restrict__ x,        // [B, 6]
    const float* __restrict__ weights,  // [3, 6] (uniform -> scalar loads)
    float* __restrict__ qout8,          // [B, 8] (6 Z-expectations + 2 zero pad)
    int B)
{
    const int b = blockIdx.x * blockDim.x + threadIdx.x;
    if (b >= B) return;

    float xv[6];
#pragma unroll
    for (int i = 0; i < 6; ++i) xv[i] = x[(size_t)b * 6 + i];

    float sr[64], si[64];
#pragma unroll
    for (int i = 0; i < 64; ++i) { sr[i] = 0.f; si[i] = 0.f; }
    sr[0] = 1.f;

    // Angle encoding: RX(x[i]) RY(x[i+1]) RZ(x[i+2]) on wire i
#pragma unroll
    for (int i = 0; i < 6; ++i) {
        {
            float t = xv[i] * 0.5f;
            rx_gate(sr, si, i, __cosf(t), __sinf(t));
        }
        {
            float t = xv[(i + 1) % 6] * 0.5f;
            ry_gate(sr, si, i, __cosf(t), __sinf(t));
        }
        {
            float t = xv[(i + 2) % 6] * 0.5f;
            rz_gate(sr, si, i, __cosf(t), __sinf(t));
        }
    }

    // Variational layers: RY per qubit + CNOT ring
#pragma unroll
    for (int l = 0; l < 3; ++l) {
#pragma unroll
        for (int q = 0; q < 6; ++q) {
            float t = weights[l * 6 + q] * 0.5f;  // wave-uniform
            ry_gate(sr, si, q, __cosf(t), __sinf(t));
        }
#pragma unroll
        for (int q = 0; q < 6; ++q) cnot_gate(sr, si, q, (q + 1) % 6);
    }

    // <Z_q> = sum_d |amp[d]|^2 * (bit_q(d) ? -1 : +1), qubit 0 = MSB
    float z[6] = {0.f, 0.f, 0.f, 0.f, 0.f, 0.f};
#pragma unroll
    for (int d = 0; d < 64; ++d) {
        float p = sr[d] * sr[d] + si[d] * si[d];
#pragma unroll
        for (int q = 0; q < 6; ++q)
            z[q] += ((d >> (5 - q)) & 1) ? -p : p;
    }

    float* o = qout8 + (size_t)b * 8;
#pragma unroll
    for (int q = 0; q < 6; ++q) o[q] = z[q];
    o[6] = 0.f;
    o[7] = 0.f;
}

// ---------------------------------------------------------------------------
// Kernel 2: out[b,h] = hx[b,h] + bias[h] + sum_{k<6} qout[b,k] * W[h,k]
// Each wave owns a 64(batch) x 16(H) strip = four 16x16 WMMA tiles, reusing
// one B fragment (fc_weight) and bias across 8 x V_WMMA_F32_16X16X4_F32.
// The hx / out streams (2 x 512 MiB, touched once) use non-temporal hints so
// they don't thrash the 192 MB L2 that caches qout8 / fcw / bias.
// H is a template parameter so the row stride (j*H*4 <= 56 KiB for H=2048)
// folds into the VMEM 24-bit immediate offset: one 64-bit base per tile,
// immediate-offset loads/stores instead of v_add_nc_u64 chains.
// ---------------------------------------------------------------------------

template <int HC>
__global__ __launch_bounds__(256, 2) void fc_wmma_kernel(
    const float* __restrict__ qout8,  // [B, 8]
    const float* __restrict__ hx,     // [B, H]
    const float* __restrict__ fcw,    // [H, 6]
    const float* __restrict__ bias,   // [H]
    float* __restrict__ out,          // [B, H]
    int H_rt)
{
    const int H = (HC > 0) ? HC : H_rt;

    const int lane = threadIdx.x & 31;
    const int wave = threadIdx.x >> 5;
    const int b0 = blockIdx.y * 64;                    // batch strip base
    const int h0 = (blockIdx.x * 8 + wave) * 16;       // H tile base (8 waves/blk)

    const int half = lane >> 4;    // 0: lanes 0-15, 1: lanes 16-31
    const int lid  = lane & 15;

    // B fragment (4x16 f32), loaded once per wave:
    // B[k][n] = fcw[(h0+n)*6 + k]; VGPR0 K = half*2, VGPR1 K = half*2+1,
    // col N = lane%16.  K >= 6 is the zero pad.
    const float* wrow = fcw + (size_t)(h0 + lid) * 6;
    v2f b_lo, b_hi;
    {
        const int k0 = half * 2;
        b_lo.x = wrow[k0 + 0];
        b_lo.y = wrow[k0 + 1];
        const int k1 = 4 + half * 2;
        b_hi.x = (k1 + 0 < 6) ? wrow[k1 + 0] : 0.f;
        b_hi.y = (k1 + 1 < 6) ? wrow[k1 + 1] : 0.f;
    }
    const float bb = bias[h0 + lid];

#pragma unroll
    for (int t = 0; t < 4; ++t) {
        const int bt = b0 + t * 16;

        // A fragment (16x4 f32): VGPR0 K = half*2, VGPR1 K = half*2+1,
        // row M = lane%16.  Two fragments cover the padded K=0..7.
        const float* qrow = qout8 + (size_t)(bt + lid) * 8;
        v2f a_lo, a_hi;
        a_lo.x = qrow[half * 2 + 0];
        a_lo.y = qrow[half * 2 + 1];
        a_hi.x = qrow[4 + half * 2 + 0];
        a_hi.y = qrow[4 + half * 2 + 1];

        // C fragment (16x16 f32, 8 VGPRs): row M = half*8 + j, col N = lane%16.
        // One base pointer per tile; j*H folds to an immediate offset for HC>0.
        const float* hbase = hx  + (size_t)(bt + half * 8) * H + (h0 + lid);
        float*       obase = out + (size_t)(bt + half * 8) * H + (h0 + lid);

        v8f c;
#pragma unroll
        for (int j = 0; j < 8; ++j)
            c[j] = __builtin_nontemporal_load(hbase + j * H) + bb;

        c = __builtin_amdgcn_wmma_f32_16x16x4_f32(
                false, a_lo, false, b_lo, (short)0, c, false, false);
        c = __builtin_amdgcn_wmma_f32_16x16x4_f32(
                false, a_hi, false, b_hi, (short)0, c, false, false);

#pragma unroll
        for (int j = 0; j < 8; ++j)
            __builtin_nontemporal_store(c[j], obase + j * H);
    }
}

// ---------------------------------------------------------------------------

extern "C" void kernel_launch(void* const* d_in, const int* in_sizes, int n_in,
                              void* d_out, int out_size, void* d_ws, size_t ws_size,
                              hipStream_t stream) {
    const float* x       = (const float*)d_in[0];  // [B, 6]
    const float* hx      = (const float*)d_in[1];  // [B, H]
    const float* weights = (const float*)d_in[2];  // [3, 6]
    const float* fcw     = (const float*)d_in[3];  // [H, 6]
    const float* bias    = (const float*)d_in[4];  // [H]
    float* out = (float*)d_out;

    const int B = in_sizes[0] / 6;   // 65536
    const int H = in_sizes[4];       // 2048

    float* qout8 = (float*)d_ws;     // [B, 8] = 2 MB scratch

    // Phase 1: per-thread state-vector simulation -> qout8
    quantum_sim_kernel<<<(B + 127) / 128, 128, 0, stream>>>(x, weights, qout8, B);

    // Phase 2: fused K=6 GEMM + bias + residual via f32 WMMA.
    // 8 waves/block, each wave owns a 64x16 output strip -> block = 64 x 128.
    dim3 grid(H / 128, B / 64);
    if (H == 2048) {
        fc_wmma_kernel<2048><<<grid, 256, 0, stream>>>(qout8, hx, fcw, bias, out, H);
    } else {
        fc_wmma_kernel<0><<<grid, 256, 0, stream>>>(qout8, hx, fcw, bias, out, H);
    }
}